// InteractionBlock_57440892617130
// MI455X (gfx1250) — compile-verified
//
#include <hip/hip_runtime.h>
#include <hip/hip_bf16.h>
#include <math.h>

typedef float v2f __attribute__((ext_vector_type(2)));
typedef float v8f __attribute__((ext_vector_type(8)));

#define N_NODES  5000
#define N_ELEM   10
#define C_DIM    128
#define N_EDGES  100000
#define N_BESSEL 8
#define RAD_H    64
#define N_L      4
#define SH_DIM   16

#define RS8       0.3535533905932738f    // 1/sqrt(8)
#define RS64      0.125f                 // 1/sqrt(64)
#define RS128     0.08838834764831845f   // 1/sqrt(128)
#define RS1280    0.027950849718747374f  // 1/sqrt(1280)
#define OUT_SCALE (RS128 / 20.0f)        // 1/(sqrt(128)*AVG_NUM_NEIGHBORS)

#define ET    16   // edges per workgroup
#define EF_LD 12   // padded stride for edge-feat tile (even -> 8B aligned v2f reads)
#define H_LD  68   // padded stride for hidden tiles (even; conflict-free WMMA A reads)

__device__ __forceinline__ int l_of_m(int m) {
  return (m == 0) ? 0 : (m < 4) ? 1 : (m < 9) ? 2 : 3;
}

__device__ __forceinline__ float silu(float v) { return v / (1.0f + __expf(-v)); }

// One K=4 fp32 WMMA step. Layouts per CDNA5 ISA 7.12.2:
//  A 16x4 : lane m=lane&15 row; VGPR0/1 = K {0,1} (lanes<16) or {2,3} (lanes>=16)
//  B 4x16 : lane n=lane&15 col; VGPR0/1 = K {0,1} / {2,3}
//  C/D    : VGPR r -> row r + 8*(lane>=16), col = lane&15
__device__ __forceinline__ v8f wmma_step(v2f a, v2f b, v8f acc) {
  return __builtin_amdgcn_wmma_f32_16x16x4_f32(false, a, false, b, (short)0, acc,
                                               false, false);
}

// ---------------------------------------------------------------- zero msg
__global__ void zero_kernel(float4* __restrict__ p, int n4) {
  int i = blockIdx.x * blockDim.x + threadIdx.x;
  if (i < n4) p[i] = make_float4(0.f, 0.f, 0.f, 0.f);
}

// ---------------------------------------------------------------- x = F @ w_up / sqrt(C)
__global__ void node_up_kernel(const float* __restrict__ F,
                               const float* __restrict__ W,
                               float* __restrict__ X) {
  const int gw   = blockIdx.x * (blockDim.x >> 5) + (threadIdx.x >> 5);
  const int lane = threadIdx.x & 31;
  const int nRowT = (N_NODES + 15) / 16;  // 313
  if (gw >= nRowT * 8) return;            // uniform per wave
  const int rowT = gw >> 3, colT = gw & 7;
  const int row0 = rowT * 16, col0 = colT * 16;
  const int half = lane >> 4, r16 = lane & 15;
  const int arow = row0 + r16;
  // Clamp (not predicate): D row M depends only on A row M; clamped tail rows
  // produce garbage that is never stored. Keeps loads branchless & in-bounds.
  const float* __restrict__ Arow =
      F + (size_t)(arow < N_NODES ? arow : N_NODES - 1) * C_DIM + 2 * half;
  const float* __restrict__ Bcol = W + col0 + r16 + (size_t)(2 * half) * C_DIM;
  v8f acc = {0.f, 0.f, 0.f, 0.f, 0.f, 0.f, 0.f, 0.f};
  for (int k = 0; k < C_DIM; k += 4) {
    v2f a = *(const v2f*)(Arow + k);                       // global_load_b64
    v2f b;
    b.x = Bcol[(size_t)k * C_DIM];
    b.y = Bcol[(size_t)(k + 1) * C_DIM];
    acc = wmma_step(a, b, acc);
  }
  for (int r = 0; r < 8; ++r) {
    int orow = row0 + r + 8 * half;
    if (orow < N_NODES) X[(size_t)orow * C_DIM + col0 + r16] = acc[r] * RS128;
  }
}

// ---------------------------------------------------------------- sc (one-hot skip)
__global__ void skip_kernel(const float* __restrict__ F,
                            const float* __restrict__ ATTR,
                            const float* __restrict__ WS,
                            float* __restrict__ SC) {
  const int n = blockIdx.x;
  const int d = threadIdx.x;  // 128
  __shared__ int s_a;
  if (d == 0) s_a = 0;
  __syncthreads();
  if (d < N_ELEM && ATTR[(size_t)n * N_ELEM + d] > 0.5f) s_a = d;  // one-hot: one writer
  __syncthreads();
  const int a = s_a;
  float sum = 0.f;
#pragma unroll 8
  for (int c = 0; c < C_DIM; ++c)
    sum += F[(size_t)n * C_DIM + c] * WS[((size_t)c * N_ELEM + a) * C_DIM + d];
  SC[(size_t)n * C_DIM + d] = sum * RS1280;
}

// ---------------------------------------------------------------- radial MLP + scatter
__global__ void __launch_bounds__(256)
radial_scatter_kernel(const float* __restrict__ EA,    // E x 16
                      const float* __restrict__ EF,    // E x 8
                      const int*   __restrict__ EIDX,  // 2 x E
                      const float* __restrict__ X,     // N x 128
                      const float* __restrict__ W1,    // 8 x 64
                      const float* __restrict__ W2,    // 64 x 64
                      const float* __restrict__ W3,    // 64 x 64
                      const float* __restrict__ W4,    // 64 x 512
                      float* __restrict__ MSG) {       // N x 2048
  __shared__ __align__(16) float s_ef[ET * EF_LD];
  __shared__ __align__(16) float s_hA[ET * H_LD];
  __shared__ __align__(16) float s_hB[ET * H_LD];
  __shared__ __align__(16) float s_tpw[ET * N_L * C_DIM];  // 16 x 512
  __shared__ __align__(16) float s_x[ET * C_DIM];          // 16 x 128
  __shared__ __align__(16) float s_ea[ET * SH_DIM];        // 16 x 16
  __shared__ int s_snd[ET];
  __shared__ int s_rcv[ET];

  const int t    = threadIdx.x;   // blockDim = 256 (8 waves)
  const int wave = t >> 5;
  const int lane = t & 31;
  const int e0   = blockIdx.x * ET;
  const int half = lane >> 4, r16 = lane & 15;

  if (t < ET) {
    int e = e0 + t;
    s_snd[t] = (e < N_EDGES) ? EIDX[e] : 0;
    s_rcv[t] = (e < N_EDGES) ? EIDX[N_EDGES + e] : 0;
  }
  __syncthreads();
  if (t < ET * N_BESSEL) {  // edge feats: 16x8 = 128 values
    int e = t >> 3, k = t & 7;
    s_ef[e * EF_LD + k] = (e0 + e < N_EDGES) ? EF[(size_t)(e0 + e) * N_BESSEL + k] : 0.f;
  }
  for (int i = t; i < ET * SH_DIM; i += 256) {  // edge attrs
    int e = i >> 4;
    s_ea[i] = (e0 + e < N_EDGES) ? EA[(size_t)(e0 + e) * SH_DIM + (i & 15)] : 0.f;
  }
  for (int i = t; i < ET * C_DIM / 4; i += 256) {  // gather x[sender], 16B loads
    int e = i >> 5, c4 = i & 31;
    float4 v = make_float4(0.f, 0.f, 0.f, 0.f);
    if (e0 + e < N_EDGES)
      v = ((const float4*)(X + (size_t)s_snd[e] * C_DIM))[c4];
    ((float4*)s_x)[i] = v;
  }
  __syncthreads();

  // --- layers 1..3 on waves 0..3 (wave-uniform branch; EXEC all-ones inside)
  if (wave < 4) {
    const int nt = wave;
    // layer 1: (16x8) @ (8x64), silu -> s_hA
    {
      v8f acc = {0.f, 0.f, 0.f, 0.f, 0.f, 0.f, 0.f, 0.f};
      for (int k = 0; k < N_BESSEL; k += 4) {
        v2f a = *(const v2f*)(&s_ef[r16 * EF_LD + k + 2 * half]);  // ds_load_b64
        v2f b;
        b.x = W1[(k + 2 * half) * RAD_H + nt * 16 + r16];
        b.y = W1[(k + 2 * half + 1) * RAD_H + nt * 16 + r16];
        acc = wmma_step(a, b, acc);
      }
      for (int r = 0; r < 8; ++r)
        s_hA[(r + 8 * half) * H_LD + nt * 16 + r16] = silu(acc[r] * RS8);
    }
    __builtin_amdgcn_s_barrier_signal(-1);
    __builtin_amdgcn_s_barrier_wait(-1);
    // layer 2: (16x64) @ (64x64), silu -> s_hB
    {
      v8f acc = {0.f, 0.f, 0.f, 0.f, 0.f, 0.f, 0.f, 0.f};
      for (int k = 0; k < RAD_H; k += 4) {
        v2f a = *(const v2f*)(&s_hA[r16 * H_LD + k + 2 * half]);
        v2f b;
        b.x = W2[(k + 2 * half) * RAD_H + nt * 16 + r16];
        b.y = W2[(k + 2 * half + 1) * RAD_H + nt * 16 + r16];
        acc = wmma_step(a, b, acc);
      }
      for (int r = 0; r < 8; ++r)
        s_hB[(r + 8 * half) * H_LD + nt * 16 + r16] = silu(acc[r] * RS64);
    }
    __builtin_amdgcn_s_barrier_signal(-1);
    __builtin_amdgcn_s_barrier_wait(-1);
    // layer 3: (16x64) @ (64x64), silu -> s_hA
    {
      v8f acc = {0.f, 0.f, 0.f, 0.f, 0.f, 0.f, 0.f, 0.f};
      for (int k = 0; k < RAD_H; k += 4) {
        v2f a = *(const v2f*)(&s_hB[r16 * H_LD + k + 2 * half]);
        v2f b;
        b.x = W3[(k + 2 * half) * RAD_H + nt * 16 + r16];
        b.y = W3[(k + 2 * half + 1) * RAD_H + nt * 16 + r16];
        acc = wmma_step(a, b, acc);
      }
      for (int r = 0; r < 8; ++r)
        s_hA[(r + 8 * half) * H_LD + nt * 16 + r16] = silu(acc[r] * RS64);
    }
  } else {
    // idle waves still participate in the two intra-phase barriers
    __builtin_amdgcn_s_barrier_signal(-1);
    __builtin_amdgcn_s_barrier_wait(-1);
    __builtin_amdgcn_s_barrier_signal(-1);
    __builtin_amdgcn_s_barrier_wait(-1);
  }
  __syncthreads();

  // --- layer 4: (16x64) @ (64x512) -> s_tpw, all 8 waves, 4 tiles each
  for (int nt = wave; nt < 32; nt += 8) {
    v8f acc = {0.f, 0.f, 0.f, 0.f, 0.f, 0.f, 0.f, 0.f};
    for (int k = 0; k < RAD_H; k += 4) {
      v2f a = *(const v2f*)(&s_hA[r16 * H_LD + k + 2 * half]);
      v2f b;
      b.x = W4[(k + 2 * half) * (N_L * C_DIM) + nt * 16 + r16];
      b.y = W4[(k + 2 * half + 1) * (N_L * C_DIM) + nt * 16 + r16];
      acc = wmma_step(a, b, acc);
    }
    for (int r = 0; r < 8; ++r)
      s_tpw[(r + 8 * half) * (N_L * C_DIM) + nt * 16 + r16] = acc[r] * RS64;
  }
  __syncthreads();

  // --- fused mji + segment_sum: atomic scatter into L2-resident message
  for (int i = t; i < ET * SH_DIM * C_DIM; i += 256) {
    int e = i >> 11;
    int rem = i & 2047;
    int m = rem >> 7;
    int c = rem & 127;
    if (e0 + e < N_EDGES) {
      float val = s_x[e * C_DIM + c] * s_ea[e * SH_DIM + m] *
                  s_tpw[e * (N_L * C_DIM) + l_of_m(m) * C_DIM + c];
      atomicAdd(&MSG[(size_t)s_rcv[e] * (SH_DIM * C_DIM) + rem], val);
    }
  }
}

// ---------------------------------------------------------------- out[n,m,:] = msg[n,m,:] @ w_out[l_of_m(m)]
__global__ void out_gemm_kernel(const float* __restrict__ MSG,   // N x 2048
                                const float* __restrict__ Wout,  // 4 x 128 x 128
                                float* __restrict__ OUT) {       // N x 2048
  const int gw   = blockIdx.x * (blockDim.x >> 5) + (threadIdx.x >> 5);
  const int lane = threadIdx.x & 31;
  const int nRowT = (N_NODES + 15) / 16;
  if (gw >= nRowT * SH_DIM * 8) return;  // uniform per wave
  const int rowT = gw >> 7;
  const int rem  = gw & 127;
  const int m    = rem >> 3;
  const int colT = rem & 7;
  const float* __restrict__ W = Wout + (size_t)l_of_m(m) * C_DIM * C_DIM;
  const int row0 = rowT * 16, col0 = colT * 16;
  const int half = lane >> 4, r16 = lane & 15;
  const int arow = row0 + r16;
  const float* __restrict__ Arow =
      MSG + (size_t)(arow < N_NODES ? arow : N_NODES - 1) * (SH_DIM * C_DIM) +
      m * C_DIM + 2 * half;
  const float* __restrict__ Bcol = W + col0 + r16 + (size_t)(2 * half) * C_DIM;
  v8f acc = {0.f, 0.f, 0.f, 0.f, 0.f, 0.f, 0.f, 0.f};
  for (int k = 0; k < C_DIM; k += 4) {
    v2f a = *(const v2f*)(Arow + k);                       // global_load_b64
    v2f b;
    b.x = Bcol[(size_t)k * C_DIM];
    b.y = Bcol[(size_t)(k + 1) * C_DIM];
    acc = wmma_step(a, b, acc);
  }
  for (int r = 0; r < 8; ++r) {
    int orow = row0 + r + 8 * half;
    if (orow < N_NODES)
      OUT[(size_t)orow * (SH_DIM * C_DIM) + m * C_DIM + col0 + r16] = acc[r] * OUT_SCALE;
  }
}

extern "C" void kernel_launch(void* const* d_in, const int* in_sizes, int n_in,
                              void* d_out, int out_size, void* d_ws, size_t ws_size,
                              hipStream_t stream) {
  const float* node_attrs = (const float*)d_in[0];
  const float* node_feats = (const float*)d_in[1];
  const float* edge_attrs = (const float*)d_in[2];
  const float* edge_feats = (const float*)d_in[3];
  const int*   edge_index = (const int*)d_in[4];
  const float* w_up   = (const float*)d_in[5];
  const float* w_rad1 = (const float*)d_in[6];
  const float* w_rad2 = (const float*)d_in[7];
  const float* w_rad3 = (const float*)d_in[8];
  const float* w_rad4 = (const float*)d_in[9];
  const float* w_skip = (const float*)d_in[10];
  const float* w_out  = (const float*)d_in[11];

  float* out = (float*)d_out;                            // N x 2048 (tuple elem 0)
  float* sc  = out + (size_t)N_NODES * SH_DIM * C_DIM;   // N x 128  (tuple elem 1)

  float* x   = (float*)d_ws;                             // N x 128
  float* msg = x + (size_t)N_NODES * C_DIM;              // N x 2048 (~41 MB)

  {  // zero message accumulator (must self-init every call)
    int n4 = (N_NODES * SH_DIM * C_DIM) / 4;
    zero_kernel<<<(n4 + 255) / 256, 256, 0, stream>>>((float4*)msg, n4);
  }
  {  // x = node_feats @ w_up / sqrt(C)
    int waves = ((N_NODES + 15) / 16) * 8;
    node_up_kernel<<<(waves + 7) / 8, 256, 0, stream>>>(node_feats, w_up, x);
  }
  // sc (independent of the rest)
  skip_kernel<<<N_NODES, C_DIM, 0, stream>>>(node_feats, node_attrs, w_skip, sc);

  // radial MLP + fused scatter (needs x and zeroed msg; stream-ordered)
  radial_scatter_kernel<<<(N_EDGES + ET - 1) / ET, 256, 0, stream>>>(
      edge_attrs, edge_feats, edge_index, x, w_rad1, w_rad2, w_rad3, w_rad4, msg);

  {  // out = msg @ w_out[l_of_m] / (sqrt(C)*avg_neighbors)
    int waves = ((N_NODES + 15) / 16) * SH_DIM * 8;
    out_gemm_kernel<<<(waves + 7) / 8, 256, 0, stream>>>(msg, w_out, out);
  }
}